// RefractiveSurface_24043226923988
// MI455X (gfx1250) — compile-verified
//
#include <hip/hip_runtime.h>

// Streaming, bandwidth-bound kernel for MI455X (gfx1250, wave32).
// 8M rays * 32 B/ray = 256 MB stream-once -> ~11 us floor at 23.3 TB/s.
// v2: replace IEEE div/sqrt expansions with CDNA hardware transcendentals
// (v_rcp_f32 + 1 NR step, v_rsq_f32, v_sqrt_f32) to halve VALU work and keep
// the kernel memory-bound. Memory path: b128 nontemporal loads/stores,
// 2 rays per lane. No WMMA: elementwise op, no matrix contraction exists.

typedef float fvec4 __attribute__((ext_vector_type(4)));

#define EPS_A 1e-12f
#define BIG   1e30f
#define ETA   (1.0f / 1.5f)   // N1/N2

// v_rcp_f32 (<=1 ulp) + one Newton-Raphson step -> ~0.5 ulp reciprocal,
// ~3 VALU instrs instead of the ~10-instr IEEE division sequence.
__device__ __forceinline__ float fast_rcp(float x) {
    float r = __builtin_amdgcn_rcpf(x);
    float e = __builtin_fmaf(-x, r, 1.0f);
    return __builtin_fmaf(r, e, r);
}

__device__ __forceinline__ float fast_sqrt(float x) {   // v_sqrt_f32, <=1 ulp
    return __builtin_amdgcn_sqrtf(x);
}

__device__ __forceinline__ float fast_rsqrt(float x) {  // v_rsq_f32, <=1 ulp
    return __builtin_amdgcn_rsqf(x);
}

__device__ __forceinline__ void trace_ray(float px, float py, float vx, float vy,
                                          float A,
                                          float& cx, float& cy,
                                          float& ox, float& oy) {
    // Quadratic A*y^2 = x intersection, numerically-stable form (matches reference).
    float a    = A * vy * vy;
    float b    = 2.0f * A * py * vy - vx;
    float c    = A * py * py - px;
    float disc = b * b - 4.0f * a * c;
    float sq   = fast_sqrt(fmaxf(disc, 0.0f));
    float sgnb = (b >= 0.0f) ? 1.0f : -1.0f;
    float q    = -0.5f * (b + sgnb * sq);

    float a_safe = (fabsf(a) < EPS_A) ? 1.0f : a;
    float q_safe = (fabsf(q) < EPS_A) ? 1.0f : q;
    float r1 = c * fast_rcp(q_safe);
    float r2 = q * fast_rcp(a_safe);
    float r1p = (r1 > 0.0f) ? r1 : BIG;
    float r2p = (r2 > 0.0f) ? r2 : BIG;
    float t_quad = fminf(r1p, r2p);

    float b_safe = (fabsf(b) < EPS_A) ? 1.0f : b;
    float t_lin  = -c * fast_rcp(b_safe);
    float t      = (fabsf(a) < EPS_A) ? t_lin : t_quad;

    cx = px + t * vx;
    cy = py + t * vy;

    // Surface normal at collision point: n ~ (1, -2*A*cy) normalized.
    float ny       = -2.0f * A * cy;
    float inv_norm = fast_rsqrt(__builtin_fmaf(ny, ny, 1.0f));
    float nx  = inv_norm;
    float nyn = ny * inv_norm;

    float dot = nx * vx + nyn * vy;
    if (dot > 0.0f) { nx = -nx; nyn = -nyn; }

    // Snell refraction with total-internal-reflection fallback.
    float cosi = -(nx * vx + nyn * vy);
    float k    = 1.0f - ETA * ETA * (1.0f - cosi * cosi);
    float cost = fast_sqrt(fmaxf(k, 0.0f));

    float s  = ETA * cosi - cost;
    float rx = ETA * vx + s * nx;     // refracted
    float ry = ETA * vy + s * nyn;
    float lx = vx + 2.0f * cosi * nx; // reflected
    float ly = vy + 2.0f * cosi * nyn;

    ox = (k < 0.0f) ? lx : rx;
    oy = (k < 0.0f) ? ly : ry;
}

// Main path: one float4 (= 2 rays) per lane. b128 NT loads/stores, fully coalesced.
__global__ __launch_bounds__(256) void refract_vec4_kernel(
        const fvec4* __restrict__ orig,   // (N/2) x [px0,py0,px1,py1]
        const fvec4* __restrict__ vec,    // (N/2) x [vx0,vy0,vx1,vy1]
        const float* __restrict__ Ap,     // scalar
        fvec4* __restrict__ cpts,         // collision points, same packing
        fvec4* __restrict__ ovec,         // out vectors, same packing
        int npairs) {
    int i = blockIdx.x * blockDim.x + threadIdx.x;
    if (i >= npairs) return;

    const float A = Ap[0];                       // uniform -> s_load
    fvec4 o = __builtin_nontemporal_load(orig + i);
    fvec4 v = __builtin_nontemporal_load(vec + i);

    float c0x, c0y, w0x, w0y;
    float c1x, c1y, w1x, w1y;
    trace_ray(o.x, o.y, v.x, v.y, A, c0x, c0y, w0x, w0y);
    trace_ray(o.z, o.w, v.z, v.w, A, c1x, c1y, w1x, w1y);

    fvec4 c = { c0x, c0y, c1x, c1y };
    fvec4 w = { w0x, w0y, w1x, w1y };

    __builtin_nontemporal_store(c, cpts + i);
    __builtin_nontemporal_store(w, ovec + i);
}

// Scalar tail for odd N (deterministic; not launched when N is even).
__global__ void refract_tail_kernel(const float* __restrict__ orig,
                                    const float* __restrict__ vec,
                                    const float* __restrict__ Ap,
                                    float* __restrict__ cpts,
                                    float* __restrict__ ovec,
                                    int start, int N) {
    int i = start + blockIdx.x * blockDim.x + threadIdx.x;
    if (i >= N) return;
    const float A = Ap[0];
    float cx, cy, ox, oy;
    trace_ray(orig[2 * i], orig[2 * i + 1], vec[2 * i], vec[2 * i + 1], A,
              cx, cy, ox, oy);
    cpts[2 * i]     = cx;
    cpts[2 * i + 1] = cy;
    ovec[2 * i]     = ox;
    ovec[2 * i + 1] = oy;
}

extern "C" void kernel_launch(void* const* d_in, const int* in_sizes, int n_in,
                              void* d_out, int out_size, void* d_ws, size_t ws_size,
                              hipStream_t stream) {
    (void)n_in; (void)d_ws; (void)ws_size; (void)out_size;

    const float* orig = (const float*)d_in[0];   // rays_origins, 2N floats
    const float* vec  = (const float*)d_in[1];   // rays_vectors, 2N floats
    const float* Ap   = (const float*)d_in[2];   // scalar A

    const int N = in_sizes[0] / 2;               // number of rays
    float* cpts = (float*)d_out;                 // collision_points: first 2N floats
    float* ovec = (float*)d_out + 2 * (size_t)N; // out_vectors: next 2N floats

    const int npairs = N / 2;                    // float4 units (2 rays each)
    if (npairs > 0) {
        const int block = 256;
        const int grid  = (npairs + block - 1) / block;
        refract_vec4_kernel<<<grid, block, 0, stream>>>(
            (const fvec4*)orig, (const fvec4*)vec, Ap,
            (fvec4*)cpts, (fvec4*)ovec, npairs);
    }
    const int tail_start = 2 * npairs;
    if (tail_start < N) {  // only when N is odd
        refract_tail_kernel<<<1, 32, 0, stream>>>(
            orig, vec, Ap, cpts, ovec, tail_start, N);
    }
}